// DiffDelRNN_33406255628323
// MI455X (gfx1250) — compile-verified
//
#include <hip/hip_runtime.h>
#include <hip/hip_bf16.h>

typedef __attribute__((ext_vector_type(2))) float v2f;
typedef __attribute__((ext_vector_type(8))) float v8f;

#define T_LEN 65536
#define N_BATCH 32
#define H_DIM 8
#define MAXD 10000

// Fast device math: single v_exp_f32 + v_rcp_f32, no IEEE division expansion.
__device__ __forceinline__ float fast_sigmoid(float x) {
    return __builtin_amdgcn_rcpf(1.0f + __expf(-x));
}
__device__ __forceinline__ float fast_tanh(float x) {
    x = fminf(15.0f, fmaxf(-15.0f, x));          // e^30 is finite in f32
    const float e = __expf(2.0f * x);
    return (e - 1.0f) * __builtin_amdgcn_rcpf(e + 1.0f);
}

// One wave32 per workgroup handles 16 batches for the whole T=65536 scan.
// Per step: gh(24x16) = Whh(24x8) @ H^T(8x16) via 4 chained
// V_WMMA_F32_16X16X4_F32 (2 M-tiles x 2 K-slabs), biases folded into C.
// The h state lives purely in low-lane registers; B fragments are built by
// lane broadcasts (ds_bpermute) — no LDS store->load on the critical path.
__launch_bounds__(32)
__global__ void gru_scan_wmma(const float* __restrict__ x,
                              const float* __restrict__ w_ih,   // (24,1)
                              const float* __restrict__ w_hh,   // (24,8)
                              const float* __restrict__ b_ih,   // (24)
                              const float* __restrict__ b_hh,   // (24)
                              const float* __restrict__ w_out,  // (1,8)
                              float* __restrict__ pre_d)        // (N,T)
{
    const int l  = threadIdx.x;      // 0..31 (wave32)
    const int lo = l & 15;
    const int hi = l >> 4;
    const int nbase = blockIdx.x * 16;
    const int n  = nbase + lo;

    // ---- loop-invariant A fragments (Whh), 16x4 f32 layout:
    //      VGPR0 = {K0 lanes0-15, K2 lanes16-31}, VGPR1 = {K1, K3}, M = lane%16
    v2f a_rz[2], a_n[2];
    for (int kc = 0; kc < 2; ++kc) {
        const int j0 = kc * 4 + 2 * hi;
        a_rz[kc].x = w_hh[lo * H_DIM + j0];
        a_rz[kc].y = w_hh[lo * H_DIM + j0 + 1];
        const int g1 = 16 + lo;              // n-gate tile rows; >=24 are padding
        a_n[kc].x = (g1 < 24) ? w_hh[g1 * H_DIM + j0]     : 0.0f;
        a_n[kc].y = (g1 < 24) ? w_hh[g1 * H_DIM + j0 + 1] : 0.0f;
    }

    // ---- C seeds (biases) + per-lane constants.
    // C/D layout: VGPR v holds row M = v + 8*hi, col N = lane%16.
    v8f c_rz, c_n;
    float wih_rz[8], wih_n[8], bih_n[8], wo[8];
    for (int v = 0; v < 8; ++v) {
        const int g = v + 8 * hi;            // 0..15: r rows then z rows
        c_rz[v]   = b_ih[g] + b_hh[g];       // r/z: both biases fold into C
        wih_rz[v] = w_ih[g];
        const int gn = 16 + v + 8 * hi;      // 16..31: n-gate rows then padding
        c_n[v]    = (gn < 24) ? b_hh[gn] : 0.0f;  // n: only b_hh (r scales it)
        wih_n[v]  = w_ih[16 + v];
        bih_n[v]  = b_ih[16 + v];
        wo[v]     = w_out[v];
    }

    float hreg[8];
    #pragma unroll
    for (int v = 0; v < 8; ++v) hreg[v] = 0.0f;

    const float* xrow  = x + (size_t)n * T_LEN;        // C==1 => x[n*T + t]
    float*       pdrow = pre_d + (size_t)n * T_LEN;

    // Software-pipelined x: one b128 load per 4 steps, issued one block early.
    float4 xcur = *(const float4*)xrow;
    for (int t = 0; t < T_LEN; t += 4) {
        const int tn = (t + 4) & (T_LEN - 1);          // wraps to 0 on last iter
        const float4 xnxt = *(const float4*)(xrow + tn);
        const float xs[4] = {xcur.x, xcur.y, xcur.z, xcur.w};
        float pd4[4];

        #pragma unroll
        for (int u = 0; u < 4; ++u) {
            const float xn = xs[u];

            // B fragments: H^T (K=8 x N=16), same K-pairing as A:
            // lane l needs hreg[kc*4 + 2*hi + c] of lane (l&15) —
            // pure register broadcast, no LDS memory traffic.
            float tb[8];
            #pragma unroll
            for (int j = 0; j < 8; ++j)
                tb[j] = __shfl(hreg[j], lo, 32);
            v2f b0, b1;
            b0.x = hi ? tb[2] : tb[0];
            b0.y = hi ? tb[3] : tb[1];
            b1.x = hi ? tb[6] : tb[4];
            b1.y = hi ? tb[7] : tb[5];

            v8f d_rz = __builtin_amdgcn_wmma_f32_16x16x4_f32(
                false, a_rz[0], false, b0, (short)0, c_rz, false, false);
            d_rz = __builtin_amdgcn_wmma_f32_16x16x4_f32(
                false, a_rz[1], false, b1, (short)0, d_rz, false, false);
            v8f d_n = __builtin_amdgcn_wmma_f32_16x16x4_f32(
                false, a_n[0], false, b0, (short)0, c_n, false, false);
            d_n = __builtin_amdgcn_wmma_f32_16x16x4_f32(
                false, a_n[1], false, b1, (short)0, d_n, false, false);

            // Gates: lanes 0-15 compute r[j], lanes 16-31 compute z[j].
            float rz[8], zsh[8];
            #pragma unroll
            for (int v = 0; v < 8; ++v)
                rz[v] = fast_sigmoid(d_rz[v] + xn * wih_rz[v]);
            #pragma unroll
            for (int v = 0; v < 8; ++v)
                zsh[v] = __shfl_down(rz[v], 16, 32);   // low lanes receive z

            // Unpredicated update: hi-lanes' d_n rows are zero-padded and all
            // shuffled values are finite, so their (unused) h stays finite.
            float pd = 0.0f;
            #pragma unroll
            for (int v = 0; v < 8; ++v) {
                const float nn = fast_tanh(xn * wih_n[v] + bih_n[v] + rz[v] * d_n[v]);
                const float hv = nn + zsh[v] * (hreg[v] - nn);  // (1-z)n + z h
                hreg[v] = hv;
                pd += hv * wo[v];
            }
            pd4[u] = pd;
        }

        if (hi == 0)
            *(float4*)(pdrow + t) = make_float4(pd4[0], pd4[1], pd4[2], pd4[3]);
        xcur = xnxt;
    }
}

// Time-varying 2-tap fractional delay: pure gather + lerp, bandwidth bound.
__global__ void delay_line_kernel(const float* __restrict__ dt,     // (N,T)
                                  const float* __restrict__ buffer, // (N,MAXD)
                                  const float* __restrict__ pre_d,  // (N,T)
                                  float* __restrict__ y)            // (N,T)
{
    const int idx = blockIdx.x * blockDim.x + threadIdx.x;
    if (idx >= N_BATCH * T_LEN) return;
    const int n = idx >> 16;            // T_LEN == 65536
    const int t = idx & (T_LEN - 1);

    const float p  = (float)MAXD - dt[idx];   // fractional tap position
    const float k0 = floorf(p);
    const float fr = p - k0;                  // in [0,1)
    const int   k0i = (int)k0;
    const int   k1i = min(k0i + 1, MAXD);     // clamp; weight is 0 there anyway

    const int j0 = t + k0i;                   // index into concat(buffer, pre_d)
    const int j1 = t + k1i;
    const float y0 = (j0 < MAXD) ? buffer[n * MAXD + j0]
                                 : pre_d[n * T_LEN + (j0 - MAXD)];
    const float y1 = (j1 < MAXD) ? buffer[n * MAXD + j1]
                                 : pre_d[n * T_LEN + (j1 - MAXD)];
    y[idx] = (1.0f - fr) * y0 + fr * y1;
}

extern "C" void kernel_launch(void* const* d_in, const int* in_sizes, int n_in,
                              void* d_out, int out_size, void* d_ws, size_t ws_size,
                              hipStream_t stream) {
    const float* x        = (const float*)d_in[0];
    const float* del_traj = (const float*)d_in[1];
    const float* buffer   = (const float*)d_in[2];
    const float* w_ih     = (const float*)d_in[3];
    const float* w_hh     = (const float*)d_in[4];
    const float* b_ih     = (const float*)d_in[5];
    const float* b_hh     = (const float*)d_in[6];
    const float* w_out    = (const float*)d_in[7];

    float* y     = (float*)d_out;                       // output 0: (N,1,T)
    float* pre_d = y + (size_t)N_BATCH * T_LEN;         // output 1: (N,1,T)

    // Stage 1: sequential GRU scan + output projection (latency-bound).
    gru_scan_wmma<<<2, 32, 0, stream>>>(x, w_ih, w_hh, b_ih, b_hh, w_out, pre_d);

    // Stage 2: delay-line gather (bandwidth-bound), reads pre_d written above.
    const int total = N_BATCH * T_LEN;
    delay_line_kernel<<<(total + 255) / 256, 256, 0, stream>>>(
        del_traj, buffer, pre_d, y);
}